// Model_62577673503907
// MI455X (gfx1250) — compile-verified
//
#include <hip/hip_runtime.h>

typedef float v2f __attribute__((ext_vector_type(2)));
typedef float v4f __attribute__((ext_vector_type(4)));
typedef float v8f __attribute__((ext_vector_type(8)));

#define CHUNK   2048         // bodies staged in LDS per pass (32 KB as float4)
#define BLOCK_T 64           // 2 waves per block
#define WAVES_PB 2
#define ROWS_PB (WAVES_PB * 16)

// One 16-j tile = 4 K-blocks of V_WMMA_F32_16X16X4_F32.
// A-fragment (16x4 f32): lane L -> row M=L&15; VGPR v, half h -> K = v+2h.
// B-fragment (4x16 f32): lane L -> col N=L&15; VGPR v, half h -> K = v+2h.
// Both fragments for lane L at (v,h) reference the SAME j = jt+4kb+2h+v, so
// B is built branchlessly from the p.xyz this lane already loaded using
// lane-constant 0/1 float masks: bfr = m0*x + m1*y + m2*z + m3 (col 3 = 1.0
// feeds the row-sum). The diagonal j==i term cancels analytically in
// accel = pos_i*S - (inv@pos), so no masking is needed in the main path.
template <bool TAIL>
__device__ __forceinline__ v8f tile16(v8f acc, const v4f* __restrict__ sp,
                                      int jt, int c0, int n, float soft3,
                                      float xi, float yi, float zi,
                                      float m0, float m1, float m2, float m3,
                                      int half) {
#pragma unroll
  for (int kb = 0; kb < 4; ++kb) {
    const int jb = jt + kb * 4 + half * 2;
    v2f a, bfr;
#pragma unroll
    for (int v = 0; v < 2; ++v) {
      const int j = jb + v;
      const v4f p = sp[j];                         // ds_load_b128
      const float dx = xi - p.x;
      const float dy = yi - p.y;
      const float dz = zi - p.z;
      const float r2 = fmaf(dx, dx, fmaf(dy, dy, fmaf(dz, dz, soft3)));
      const float rs = __builtin_amdgcn_rsqf(r2);  // v_rsq_f32
      float inv = (rs * rs) * rs;                  // r2^-1.5 (G applied at end)
      if (TAIL) {
        inv = ((c0 + j) >= n) ? 0.f : inv;         // zero OOB tail columns
      }
      a[v] = inv;
      bfr[v] = fmaf(m0, p.x, fmaf(m1, p.y, fmaf(m2, p.z, m3)));
    }
    acc = __builtin_amdgcn_wmma_f32_16x16x4_f32(
        false, a, false, bfr, (short)0, acc, false, false);
  }
  return acc;
}

__global__ __launch_bounds__(BLOCK_T)
void nbody_wmma(const float* __restrict__ space,
                const float* __restrict__ Gp,
                const float* __restrict__ dtp,
                const float* __restrict__ softp,
                float* __restrict__ out, int n) {
  __shared__ v4f  sp[CHUNK];               // staged positions (x,y,z,1)
  __shared__ float sred[WAVES_PB][16][4];  // D-fragment spill: cols 0..3

  const float G     = Gp[0];
  const float dt    = dtp[0];
  const float soft3 = 3.0f * softp[0];     // softening added per component, d=3

  const int tid  = threadIdx.x;
  const int wave = tid >> 5;
  const int lane = tid & 31;
  const int half = lane >> 4;              // 0: lanes 0-15, 1: lanes 16-31
  const int l15  = lane & 15;

  // Lane-constant B-column masks (branchless select): col n<3 -> pos comp,
  // col 3 -> 1.0 (row-sum), cols 4..15 -> 0.
  const float m0 = (l15 == 0) ? 1.f : 0.f;
  const float m1 = (l15 == 1) ? 1.f : 0.f;
  const float m2 = (l15 == 2) ? 1.f : 0.f;
  const float m3 = (l15 == 3) ? 1.f : 0.f;

  const int rbase  = blockIdx.x * ROWS_PB + wave * 16;  // wave's 16-row base
  const int i      = rbase + l15;                       // my body row
  const int iclamp = (i < n) ? i : (n - 1);

  // space layout: [i][d][2] -> pos at i*6 + d*2, vel at i*6 + d*2 + 1
  const float xi = space[iclamp * 6 + 0];
  const float yi = space[iclamp * 6 + 2];
  const float zi = space[iclamp * 6 + 4];

  v8f acc = {0.f, 0.f, 0.f, 0.f, 0.f, 0.f, 0.f, 0.f};

  for (int c0 = 0; c0 < n; c0 += CHUNK) {
    __syncthreads();  // previous chunk fully consumed before overwrite
    for (int b = tid; b < CHUNK; b += BLOCK_T) {
      const int g  = c0 + b;
      const bool ok = (g < n);
      v4f p;
      p.x = ok ? space[g * 6 + 0] : 0.f;   // zero-fill keeps WMMA NaN-free
      p.y = ok ? space[g * 6 + 2] : 0.f;
      p.z = ok ? space[g * 6 + 4] : 0.f;
      p.w = 1.0f;
      sp[b] = p;
    }
    __syncthreads();

    for (int jt = 0; jt < CHUNK; jt += 16) {
      // Tail condition has no threadIdx dependence -> uniform scalar branch.
      if (c0 + jt + 16 > n) {
        acc = tile16<true >(acc, sp, jt, c0, n, soft3, xi, yi, zi,
                            m0, m1, m2, m3, half);
      } else {
        acc = tile16<false>(acc, sp, jt, c0, n, soft3, xi, yi, zi,
                            m0, m1, m2, m3, half);
      }
    }
  }

  // Spill D cols 0..3 (lane L holds column N=L&15, rows M = r + 8*half)
  if (l15 < 4) {
#pragma unroll
    for (int r = 0; r < 8; ++r) {
      sred[wave][r + half * 8][l15] = acc[r];
    }
  }
  __syncthreads();

  // Lanes 0..15 of each wave: Euler update for their row.
  // accel = G * (pos*S - inv@pos); diagonal term cancels in the subtraction.
  if (half == 0) {
    const int row = rbase + lane;
    if (row < n) {
      const float S  = sred[wave][lane][3];
      const float d0 = sred[wave][lane][0];
      const float d1 = sred[wave][lane][1];
      const float d2 = sred[wave][lane][2];
      const float px = space[row * 6 + 0], vx0 = space[row * 6 + 1];
      const float py = space[row * 6 + 2], vy0 = space[row * 6 + 3];
      const float pz = space[row * 6 + 4], vz0 = space[row * 6 + 5];
      const float ax = G * (px * S - d0);
      const float ay = G * (py * S - d1);
      const float az = G * (pz * S - d2);
      const float vx = vx0 - dt * ax;
      const float vy = vy0 - dt * ay;
      const float vz = vz0 - dt * az;
      out[row * 6 + 0] = px + dt * vx;  out[row * 6 + 1] = vx;
      out[row * 6 + 2] = py + dt * vy;  out[row * 6 + 3] = vy;
      out[row * 6 + 4] = pz + dt * vz;  out[row * 6 + 5] = vz;
    }
  }
}

extern "C" void kernel_launch(void* const* d_in, const int* in_sizes, int n_in,
                              void* d_out, int out_size, void* d_ws, size_t ws_size,
                              hipStream_t stream) {
  const float* space = (const float*)d_in[0];
  const float* G     = (const float*)d_in[1];
  const float* dt    = (const float*)d_in[2];
  const float* soft  = (const float*)d_in[3];
  float* out = (float*)d_out;

  const int n = in_sizes[0] / 6;            // (n, 3, 2) fp32
  const int blocks = (n + ROWS_PB - 1) / ROWS_PB;
  nbody_wmma<<<blocks, BLOCK_T, 0, stream>>>(space, G, dt, soft, out, n);
}